// Model_89154931130490
// MI455X (gfx1250) — compile-verified
//
#include <hip/hip_runtime.h>

// ---------------------------------------------------------------------------
// CDNA5 (gfx1250) wave32 WMMA fp32 implementation of the reference model.
// All matrix math uses V_WMMA_F32_16X16X4_F32 (fp32 matrix pipe).
// ---------------------------------------------------------------------------

typedef __attribute__((ext_vector_type(2))) float v2f;
typedef __attribute__((ext_vector_type(8))) float v8f;

__device__ __forceinline__ v8f wmma4(v2f a, v2f b, v8f c) {
  // D = A(16x4) * B(4x16) + C(16x16), fp32
  return __builtin_amdgcn_wmma_f32_16x16x4_f32(false, a, false, b, (short)0, c,
                                               false, false);
}

// sigmoid via native v_exp_f32 + v_rcp_f32 (avoids IEEE div expansion)
__device__ __forceinline__ float sigmoidf_(float x) {
  return __builtin_amdgcn_rcpf(1.f + __expf(-x));
}

// ---------------------------------------------------------------------------
// Generic fp32 WMMA GEMM:  C = epilogue(alpha * A@B + bias)
//   A: MxK row-major (stride lda)
//   B: element B[k][n] = Bp[k*ldbK + n*ldbN]   (supports transposed access)
//   C: element C[m][n] = Cp[m*ldcM + n*ldcN]
//   epilogue: v = alpha*acc + bias[n]; act(v); v *= aux[m,n]; v += addScale*add[m,n]
//   act: 0=none 1=lrelu(slope) 2=sigmoid 3=tanh
// One wave computes a 16 x (NT*16) tile (A fetched once, reused NT times).
// Guard-free main loop; KTAIL adds one guarded step (K%4!=0); NGUARD guards
// partial last n-tiles (N%16!=0). M must be a multiple of 16 (true here).
// EXEC is full for all WMMA ops.
// ---------------------------------------------------------------------------
template <int NT, bool KTAIL, bool NGUARD>
__global__ __launch_bounds__(128) void gemm_wmma(
    const float* __restrict__ A, int lda, const float* __restrict__ B,
    long long ldbK, long long ldbN, const float* __restrict__ bias,
    float* __restrict__ C, long long ldcM, long long ldcN,
    const float* __restrict__ aux, const float* __restrict__ addm,
    long long ldAux, int M, int N, int K, float alpha, float addScale, int act,
    float slope) {
  int wave = threadIdx.x >> 5;
  int lane = threadIdx.x & 31;
  int mt = blockIdx.y * 4 + wave;
  int m0 = mt * 16;
  int n0 = blockIdx.x * (NT * 16);
  if (m0 >= M) return;  // whole-wave uniform exit

  int hi = lane >> 4;  // 0 or 1
  int l15 = lane & 15;
  int kb = hi * 2;     // A/B K-sub-offset per ISA layout
  const float* Ar = A + (long long)(m0 + l15) * lda;

  // per-tile B column pointers + validity
  const float* Bc[NT];
  bool bval[NT];
#pragma unroll
  for (int t = 0; t < NT; ++t) {
    int bcol = n0 + t * 16 + l15;
    bval[t] = NGUARD ? (bcol < N) : true;
    Bc[t] = B + (long long)(NGUARD ? (bval[t] ? bcol : 0) : bcol) * ldbN;
  }

  v8f acc[NT];
#pragma unroll
  for (int t = 0; t < NT; ++t)
    acc[t] = (v8f){0.f, 0.f, 0.f, 0.f, 0.f, 0.f, 0.f, 0.f};

  int Kmain = KTAIL ? (K & ~3) : K;
  for (int k0 = 0; k0 < Kmain; k0 += 4) {
    int ka0 = k0 + kb;
    v2f a;
    a.x = Ar[ka0];
    a.y = Ar[ka0 + 1];
#pragma unroll
    for (int t = 0; t < NT; ++t) {
      v2f b;
      b.x = Bc[t][(long long)ka0 * ldbK];
      b.y = Bc[t][(long long)(ka0 + 1) * ldbK];
      acc[t] = wmma4(a, b, acc[t]);
    }
  }
  if (KTAIL && Kmain < K) {
    int ka0 = Kmain + kb;
    v2f a;
    a.x = (ka0 < K) ? Ar[ka0] : 0.f;
    a.y = (ka0 + 1 < K) ? Ar[ka0 + 1] : 0.f;
#pragma unroll
    for (int t = 0; t < NT; ++t) {
      v2f b;
      b.x = (ka0 < K) ? Bc[t][(long long)ka0 * ldbK] : 0.f;
      b.y = (ka0 + 1 < K) ? Bc[t][(long long)(ka0 + 1) * ldbK] : 0.f;
      acc[t] = wmma4(a, b, acc[t]);
    }
  }

#pragma unroll
  for (int t = 0; t < NT; ++t) {
    int bcol = n0 + t * 16 + l15;
    if (NGUARD && !bval[t]) continue;
    float bv = bias ? bias[bcol] : 0.f;
#pragma unroll
    for (int r = 0; r < 8; ++r) {
      int mrow = m0 + r + 8 * hi;
      float v = alpha * acc[t][r] + bv;
      if (act == 1)
        v = v > 0.f ? v : slope * v;
      else if (act == 2)
        v = sigmoidf_(v);
      else if (act == 3)
        v = tanhf(v);
      if (aux) v *= aux[(long long)mrow * ldAux + bcol];
      if (addm) v += addScale * addm[(long long)mrow * ldAux + bcol];
      C[(long long)mrow * ldcM + (long long)bcol * ldcN] = v;
    }
  }
}

// ---------------------------------------------------------------------------
// Fused spectral-block kernel: per 16-row tile (one wave):
//   R0(16x4) -> gate0 -> 16x64 -> gate1 -> gate2 -> fold(+img) -> p1W/p1b ->
//   p2W/p2b -> mval (scalar per row).  All matmuls via fp32 WMMA; LDS
//   ping-pong buffers, padded stride 66 (conflict-free A-operand reads).
// ---------------------------------------------------------------------------
__device__ __forceinline__ void gated_layer(const float* s_in, float* s_out,
                                            const float* __restrict__ W1,
                                            const float* __restrict__ B1,
                                            const float* __restrict__ W2,
                                            const float* __restrict__ B2,
                                            int hi, int l15, int kb) {
  for (int jt = 0; jt < 4; ++jt) {
    int col = jt * 16 + l15;
    v8f acc1 = {0.f, 0.f, 0.f, 0.f, 0.f, 0.f, 0.f, 0.f};
    v8f acc2 = {0.f, 0.f, 0.f, 0.f, 0.f, 0.f, 0.f, 0.f};
    for (int k0 = 0; k0 < 64; k0 += 4) {
      v2f a, b1v, b2v;
      a.x = s_in[l15 * 66 + k0 + kb];
      a.y = s_in[l15 * 66 + k0 + kb + 1];
      b1v.x = W1[(k0 + kb) * 64 + col];
      b1v.y = W1[(k0 + kb + 1) * 64 + col];
      b2v.x = W2[(k0 + kb) * 64 + col];
      b2v.y = W2[(k0 + kb + 1) * 64 + col];
      acc1 = wmma4(a, b1v, acc1);
      acc2 = wmma4(a, b2v, acc2);
    }
    float bb1 = B1[col], bb2 = B2[col];
#pragma unroll
    for (int r = 0; r < 8; ++r) {
      int row = r + 8 * hi;
      float u = acc1[r] + bb1;
      float g = acc2[r] + bb2;
      s_out[row * 66 + col] = u * sigmoidf_(g);
    }
  }
}

__global__ __launch_bounds__(128) void block_fused(
    const float* __restrict__ spec,  // (B*N) x 4, row-major
    const float* __restrict__ g0W1, const float* __restrict__ g0b1,
    const float* __restrict__ g0W2, const float* __restrict__ g0b2,
    const float* __restrict__ g1W1, const float* __restrict__ g1b1,
    const float* __restrict__ g1W2, const float* __restrict__ g1b2,
    const float* __restrict__ g2W1, const float* __restrict__ g2b1,
    const float* __restrict__ g2W2, const float* __restrict__ g2b2,
    const float* __restrict__ i3sum,  // 16 (img-path fold, constant)
    const float* __restrict__ p1W, const float* __restrict__ p1b,
    const float* __restrict__ p2W, const float* __restrict__ p2b,
    float* __restrict__ mval) {
  __shared__ float smem[4][2][16 * 66];
  int wave = threadIdx.x >> 5;
  int lane = threadIdx.x & 31;
  int m0 = (blockIdx.x * 4 + wave) * 16;
  int hi = lane >> 4, l15 = lane & 15, kb = 2 * hi;
  float* s0 = &smem[wave][0][0];
  float* s1 = &smem[wave][1][0];

  // gate0: K = 4 (single WMMA step per gate)
  v2f a0;
  a0.x = spec[(long long)(m0 + l15) * 4 + kb];
  a0.y = spec[(long long)(m0 + l15) * 4 + kb + 1];
  for (int jt = 0; jt < 4; ++jt) {
    int col = jt * 16 + l15;
    v2f b1v, b2v;
    b1v.x = g0W1[kb * 64 + col];
    b1v.y = g0W1[(kb + 1) * 64 + col];
    b2v.x = g0W2[kb * 64 + col];
    b2v.y = g0W2[(kb + 1) * 64 + col];
    v8f acc1 = {0.f, 0.f, 0.f, 0.f, 0.f, 0.f, 0.f, 0.f};
    v8f acc2 = {0.f, 0.f, 0.f, 0.f, 0.f, 0.f, 0.f, 0.f};
    acc1 = wmma4(a0, b1v, acc1);
    acc2 = wmma4(a0, b2v, acc2);
    float bb1 = g0b1[col], bb2 = g0b2[col];
#pragma unroll
    for (int r = 0; r < 8; ++r) {
      int row = r + 8 * hi;
      s0[row * 66 + col] = (acc1[r] + bb1) * sigmoidf_(acc2[r] + bb2);
    }
  }
  gated_layer(s0, s1, g1W1, g1b1, g1W2, g1b2, hi, l15, kb);
  gated_layer(s1, s0, g2W1, g2b1, g2W2, g2b2, hi, l15, kb);

  // fold: hf[t] = sum_k real[16k+t] + i3sum[t];  b0 = hf@p1W+p1b;
  // mval = b0@p2W + p2b
  if (lane < 16) {
    const float* gr = s0 + lane * 66;
    float hf[16];
#pragma unroll
    for (int t = 0; t < 16; ++t)
      hf[t] = i3sum[t] + gr[t] + gr[16 + t] + gr[32 + t] + gr[48 + t];
    float mv = p2b[0];
    for (int o = 0; o < 16; ++o) {
      float s = p1b[o];
#pragma unroll
      for (int t = 0; t < 16; ++t) s += hf[t] * p1W[t * 16 + o];
      mv += s * p2W[o];
    }
    mval[(long long)m0 + lane] = mv;
  }
}

// img path is constant (input zeros -> biases only); 64 threads.
__global__ void imgpath_kernel(
    const float* g0b1, const float* g0b2, const float* g1W1, const float* g1b1,
    const float* g1W2, const float* g1b2, const float* g2W1, const float* g2b1,
    const float* g2W2, const float* g2b2, float* i3sum) {
  __shared__ float i1[64], i2[64], i3[64];
  int t = threadIdx.x;
  i1[t] = g0b1[t] * sigmoidf_(g0b2[t]);
  __syncthreads();
  float s1 = g1b1[t], s2 = g1b2[t];
  for (int k = 0; k < 64; ++k) {
    s1 += i1[k] * g1W1[k * 64 + t];
    s2 += i1[k] * g1W2[k * 64 + t];
  }
  i2[t] = s1 * sigmoidf_(s2);
  __syncthreads();
  s1 = g2b1[t];
  s2 = g2b2[t];
  for (int k = 0; k < 64; ++k) {
    s1 += i2[k] * g2W1[k * 64 + t];
    s2 += i2[k] * g2W2[k * 64 + t];
  }
  i3[t] = s1 * sigmoidf_(s2);
  __syncthreads();
  if (t < 16) i3sum[t] = i3[t] + i3[16 + t] + i3[32 + t] + i3[48 + t];
}

// sums of wk / wq (each 256)
__global__ void sumw_kernel(const float* wk, const float* wq, float* sw) {
  int lane = threadIdx.x & 31, wave = threadIdx.x >> 5;
  const float* w = wave ? wq : wk;
  float s = 0.f;
  for (int t = lane; t < 256; t += 32) s += w[t];
#pragma unroll
  for (int off = 16; off; off >>= 1) s += __shfl_xor(s, off);
  if (lane == 0) sw[wave] = s;
}

// per-batch min / 1/(max-min+eps) of ip rows
__global__ __launch_bounds__(256) void ipnorm_kernel(const float* ip, float* mn,
                                                     float* inv) {
  int wave = threadIdx.x >> 5, lane = threadIdx.x & 31;
  int b = blockIdx.x * 8 + wave;
  const float* row = ip + (long long)b * 256;
  float lo = 3.4e38f, hv = -3.4e38f;
  for (int t = lane; t < 256; t += 32) {
    float v = row[t];
    lo = fminf(lo, v);
    hv = fmaxf(hv, v);
  }
#pragma unroll
  for (int off = 16; off; off >>= 1) {
    lo = fminf(lo, __shfl_xor(lo, off));
    hv = fmaxf(hv, __shfl_xor(hv, off));
  }
  if (lane == 0) {
    mn[b] = lo;
    inv[b] = 1.f / (hv - lo + 1e-7f);
  }
}

// attention[i,j] = mean_b softmax_j(lrelu(ki+qj,0.2)) * ipn[b,j]
// one wave per row i, batch loop inside; deterministic (no atomics).
__global__ __launch_bounds__(256) void attn_kernel(
    const float* __restrict__ x2, const float* __restrict__ ip,
    const float* __restrict__ sw, const float* __restrict__ mn,
    const float* __restrict__ inv, float* __restrict__ attn) {
  int wave = threadIdx.x >> 5, lane = threadIdx.x & 31;
  int i = blockIdx.x * 8 + wave;
  float swk = sw[0], swq = sw[1];
  float acc[8];
#pragma unroll
  for (int t = 0; t < 8; ++t) acc[t] = 0.f;
  for (int b = 0; b < 1024; ++b) {
    const float* row = x2 + (long long)b * 256;
    float ki = row[i] * swk;
    float d[8];
    float mx = -3.4e38f;
#pragma unroll
    for (int t = 0; t < 8; ++t) {
      float v = ki + row[lane + 32 * t] * swq;
      v = v > 0.f ? v : 0.2f * v;
      d[t] = v;
      mx = fmaxf(mx, v);
    }
#pragma unroll
    for (int off = 16; off; off >>= 1) mx = fmaxf(mx, __shfl_xor(mx, off));
    float s = 0.f;
#pragma unroll
    for (int t = 0; t < 8; ++t) {
      d[t] = __expf(d[t] - mx);
      s += d[t];
    }
#pragma unroll
    for (int off = 16; off; off >>= 1) s += __shfl_xor(s, off);
    float is = __builtin_amdgcn_rcpf(s);
    float bm = mn[b], bw = inv[b];
    const float* iprow = ip + (long long)b * 256;
#pragma unroll
    for (int t = 0; t < 8; ++t)
      acc[t] += d[t] * is * ((iprow[lane + 32 * t] - bm) * bw);
  }
#pragma unroll
  for (int t = 0; t < 8; ++t)
    attn[(long long)i * 256 + lane + 32 * t] = acc[t] * (1.f / 1024.f);
}

__global__ __launch_bounds__(256) void degree_kernel(const float* attn,
                                                     float* deg, float* dinv) {
  int wave = threadIdx.x >> 5, lane = threadIdx.x & 31;
  int i = blockIdx.x * 8 + wave;
  float s = 0.f;
  for (int t = lane; t < 256; t += 32) s += attn[(long long)i * 256 + t];
#pragma unroll
  for (int off = 16; off; off >>= 1) s += __shfl_xor(s, off);
  if (lane == 0) {
    deg[i] = s;
    dinv[i] = 1.f / (sqrtf(s) + 1e-7f);
  }
}

__global__ void lap_kernel(const float* attn, const float* deg,
                           const float* dinv, float* lap) {
  int i = blockIdx.x, j = threadIdx.x;
  float a = 0.5f * (attn[(long long)i * 256 + j] + attn[(long long)j * 256 + i]);
  float v = ((i == j) ? deg[i] : 0.f) - a;
  lap[(long long)i * 256 + j] = dinv[i] * v * dinv[j];
}

__global__ void zerospec_kernel(float* spec) {
  long long m = (long long)blockIdx.x * 256 + threadIdx.x;
  spec[m * 4] = 0.f;  // S_0 = 0 (mul_L[0] is the zero matrix)
}

__global__ void bn_kernel(const float* f, const float* gamma, const float* beta,
                          const float* mean, const float* var, float* y) {
  long long idx = (long long)blockIdx.x * 256 + threadIdx.x;
  int n = (int)(idx & 255);
  float v = (f[idx] - mean[n]) * rsqrtf(var[n] + 1e-5f);
  y[idx] = gamma[n] * v + beta[n];
}

__global__ __launch_bounds__(256) void prob_kernel(const float* H,
                                                   const float* W2,
                                                   const float* b2,
                                                   float* prob) {
  int wave = threadIdx.x >> 5, lane = threadIdx.x & 31;
  int b = blockIdx.x * 8 + wave;
  const float* row = H + (long long)b * 256;
  float s = 0.f;
  for (int t = lane; t < 256; t += 32) s += row[t] * W2[t];
#pragma unroll
  for (int off = 16; off; off >>= 1) s += __shfl_xor(s, off);
  if (lane == 0) prob[b] = sigmoidf_(s + b2[0]);
}

// ---------------------------------------------------------------------------
// Host side
// ---------------------------------------------------------------------------
// variant: 0 = NT4 clean (K%4==0, N%16==0), 1 = NT4 + K tail, 2 = NT3 + Nguard
static void launch_gemm(hipStream_t st, int variant, const float* A, int lda,
                        const float* B, long long ldbK, long long ldbN,
                        const float* bias, float* C, long long ldcM,
                        long long ldcN, const float* aux, const float* addm,
                        long long ldAux, int M, int N, int K, float alpha,
                        float addScale, int act, float slope) {
  dim3 blk(128);
  if (variant == 0) {
    dim3 grid((N + 63) / 64, (M + 63) / 64);
    gemm_wmma<4, false, false><<<grid, blk, 0, st>>>(
        A, lda, B, ldbK, ldbN, bias, C, ldcM, ldcN, aux, addm, ldAux, M, N, K,
        alpha, addScale, act, slope);
  } else if (variant == 1) {
    dim3 grid((N + 63) / 64, (M + 63) / 64);
    gemm_wmma<4, true, false><<<grid, blk, 0, st>>>(
        A, lda, B, ldbK, ldbN, bias, C, ldcM, ldcN, aux, addm, ldAux, M, N, K,
        alpha, addScale, act, slope);
  } else {
    dim3 grid((N + 47) / 48, (M + 63) / 64);
    gemm_wmma<3, false, true><<<grid, blk, 0, st>>>(
        A, lda, B, ldbK, ldbN, bias, C, ldcM, ldcN, aux, addm, ldAux, M, N, K,
        alpha, addScale, act, slope);
  }
}

// Input flattening assumption: recursive insertion order of setup_inputs().
enum {
  IN_X = 0, IN_IP, IN_FC_W1, IN_FC_B1, IN_FC_W2, IN_FC_B2, IN_WK, IN_WQ,
  IN_BLK0 = 8,   // 22 tensors per block
  IN_BLK1 = 30,
  IN_BN_GAMMA = 52, IN_BN_BETA, IN_BN_MEAN, IN_BN_VAR,
  IN_FP_W1 = 56, IN_FP_B1, IN_FP_W2, IN_FP_B2,
  IN_IFC_W1 = 60, IN_IFC_B1, IN_IFC_W2, IN_IFC_B2
};
enum {  // block-relative offsets
  BG0W1 = 0, BG0B1, BG0W2, BG0B2, BG1W1, BG1B1, BG1W2, BG1B2,
  BG2W1, BG2B1, BG2W2, BG2B2, BP1W, BP1B, BP2W, BP2B,
  BL1W, BL1B, BL2W, BL2B, BAW, BAB
};

// workspace layout (floats)
static const long long OFF_X2 = 0;                       // 2048*256
static const long long OFF_H = OFF_X2 + 2048LL * 256;    // 2048*256
static const long long OFF_ATTN = OFF_H + 2048LL * 256;  // 256*256
static const long long OFF_LAP = OFF_ATTN + 256LL * 256;
static const long long OFF_THIRD = OFF_LAP + 256LL * 256;
static const long long OFF_FORTH = OFF_THIRD + 256LL * 256;
static const long long OFF_SPEC = OFF_FORTH + 256LL * 256;  // 262144*4
static const long long OFF_MVAL = OFF_SPEC + 262144LL * 4;
static const long long OFF_L1 = OFF_MVAL + 262144LL;
static const long long OFF_PROD = OFF_L1 + 262144LL;
static const long long OFF_Y = OFF_PROD + 262144LL;
static const long long OFF_HFC = OFF_Y + 262144LL;
static const long long OFF_H2 = OFF_HFC + 262144LL;
static const long long OFF_DEG = OFF_H2 + 262144LL;
static const long long OFF_DINV = OFF_DEG + 256;
static const long long OFF_SW = OFF_DINV + 256;
static const long long OFF_MN = OFF_SW + 8;
static const long long OFF_INV = OFF_MN + 1024;
static const long long OFF_I3 = OFF_INV + 1024;

extern "C" void kernel_launch(void* const* d_in, const int* in_sizes, int n_in,
                              void* d_out, int out_size, void* d_ws,
                              size_t ws_size, hipStream_t stream) {
  (void)in_sizes; (void)n_in; (void)out_size; (void)ws_size;
  auto in = [&](int i) -> const float* { return (const float*)d_in[i]; };
  float* W = (float*)d_ws;
  float* out = (float*)d_out;

  float* X2x = W + OFF_X2;           // x branch (1024x256)
  float* X2p = W + OFF_X2 + 262144;  // input_prob branch (1024x256)
  float* Hx = W + OFF_H;
  float* Hp = W + OFF_H + 262144;
  float* attn = W + OFF_ATTN;
  float* lap = W + OFF_LAP;
  float* third = W + OFF_THIRD;
  float* forth = W + OFF_FORTH;
  float* spec = W + OFF_SPEC;
  float* mval = W + OFF_MVAL;
  float* l1 = W + OFF_L1;
  float* prod = W + OFF_PROD;
  float* Y = W + OFF_Y;
  float* Hfc = W + OFF_HFC;
  float* H2 = W + OFF_H2;
  float* deg = W + OFF_DEG;
  float* dinv = W + OFF_DINV;
  float* sw = W + OFF_SW;
  float* mnb = W + OFF_MN;
  float* invb = W + OFF_INV;
  float* i3sum = W + OFF_I3;

  float* prob_out = out;              // (1024)
  float* forecast = out + 1024;       // feature == forecast (1024x256)
  float* out3 = out + 1024 + 262144;  // (1024x41)

  // --- fc_shape on x and input_prob (shared weights), K=41 -> tail variant ---
  launch_gemm(stream, 1, in(IN_X), 41, in(IN_FC_W1), 256, 1, in(IN_FC_B1), Hx,
              256, 1, nullptr, nullptr, 256, 1024, 256, 41, 1.f, 0.f, 1, 0.01f);
  launch_gemm(stream, 1, in(IN_IP), 41, in(IN_FC_W1), 256, 1, in(IN_FC_B1), Hp,
              256, 1, nullptr, nullptr, 256, 1024, 256, 41, 1.f, 0.f, 1, 0.01f);
  launch_gemm(stream, 0, Hx, 256, in(IN_FC_W2), 256, 1, in(IN_FC_B2), X2x, 256,
              1, nullptr, nullptr, 256, 1024, 256, 256, 1.f, 0.f, 0, 0.f);
  launch_gemm(stream, 0, Hp, 256, in(IN_FC_W2), 256, 1, in(IN_FC_B2), X2p, 256,
              1, nullptr, nullptr, 256, 1024, 256, 256, 1.f, 0.f, 0, 0.f);

  // --- attention -> Laplacian powers ---
  sumw_kernel<<<1, 64, 0, stream>>>(in(IN_WK), in(IN_WQ), sw);
  ipnorm_kernel<<<128, 256, 0, stream>>>(X2p, mnb, invb);
  attn_kernel<<<32, 256, 0, stream>>>(X2x, X2p, sw, mnb, invb, attn);
  degree_kernel<<<32, 256, 0, stream>>>(attn, deg, dinv);
  lap_kernel<<<256, 256, 0, stream>>>(attn, deg, dinv, lap);
  launch_gemm(stream, 0, lap, 256, lap, 256, 1, nullptr, third, 256, 1, nullptr,
              nullptr, 256, 256, 256, 256, 2.f, 0.f, 0, 0.f);
  launch_gemm(stream, 0, lap, 256, third, 256, 1, nullptr, forth, 256, 1,
              nullptr, lap, 256, 256, 256, 256, 2.f, -1.f, 0, 0.f);

  // --- spectral: spec[(b*256+n)*4 + k] = sum_t x2[b,t] * Lk[n,t] ---
  zerospec_kernel<<<1024, 256, 0, stream>>>(spec);
  launch_gemm(stream, 0, X2x, 256, lap, 1, 256, nullptr, spec + 1, 1024, 4,
              nullptr, nullptr, 256, 1024, 256, 256, 1.f, 0.f, 0, 0.f);
  launch_gemm(stream, 0, X2x, 256, third, 1, 256, nullptr, spec + 2, 1024, 4,
              nullptr, nullptr, 256, 1024, 256, 256, 1.f, 0.f, 0, 0.f);
  launch_gemm(stream, 0, X2x, 256, forth, 1, 256, nullptr, spec + 3, 1024, 4,
              nullptr, nullptr, 256, 1024, 256, 256, 1.f, 0.f, 0, 0.f);

  // --- two spectral blocks ---
  for (int bi = 0; bi < 2; ++bi) {
    int base = (bi == 0) ? IN_BLK0 : IN_BLK1;
    imgpath_kernel<<<1, 64, 0, stream>>>(
        in(base + BG0B1), in(base + BG0B2), in(base + BG1W1), in(base + BG1B1),
        in(base + BG1W2), in(base + BG1B2), in(base + BG2W1), in(base + BG2B1),
        in(base + BG2W2), in(base + BG2B2), i3sum);
    block_fused<<<4096, 128, 0, stream>>>(
        spec, in(base + BG0W1), in(base + BG0B1), in(base + BG0W2),
        in(base + BG0B2), in(base + BG1W1), in(base + BG1B1), in(base + BG1W2),
        in(base + BG1B2), in(base + BG2W1), in(base + BG2B1), in(base + BG2W2),
        in(base + BG2B2), i3sum, in(base + BP1W), in(base + BP1B),
        in(base + BP2W), in(base + BP2B), mval);
    // mask: l1 = tanh(m@L1W+b); prod = sigmoid(m@L2W+b)*l1;
    //       blk = m * sigmoid(prod@AW+b)   (accumulated into forecast)
    launch_gemm(stream, 0, mval, 256, in(base + BL1W), 256, 1, in(base + BL1B),
                l1, 256, 1, nullptr, nullptr, 256, 1024, 256, 256, 1.f, 0.f, 3,
                0.f);
    launch_gemm(stream, 0, mval, 256, in(base + BL2W), 256, 1, in(base + BL2B),
                prod, 256, 1, l1, nullptr, 256, 1024, 256, 256, 1.f, 0.f, 2,
                0.f);
    launch_gemm(stream, 0, prod, 256, in(base + BAW), 256, 1, in(base + BAB),
                forecast, 256, 1, mval, (bi ? forecast : nullptr), 256, 1024,
                256, 256, 1.f, 1.f, 2, 0.f);
  }

  // --- BN + fc_prob ---
  bn_kernel<<<1024, 256, 0, stream>>>(forecast, in(IN_BN_GAMMA),
                                      in(IN_BN_BETA), in(IN_BN_MEAN),
                                      in(IN_BN_VAR), Y);
  launch_gemm(stream, 0, Y, 256, in(IN_FP_W1), 256, 1, in(IN_FP_B1), Hfc, 256,
              1, nullptr, nullptr, 256, 1024, 256, 256, 1.f, 0.f, 1, 0.01f);
  prob_kernel<<<128, 256, 0, stream>>>(Hfc, in(IN_FP_W2), in(IN_FP_B2),
                                       prob_out);

  // --- ifc ---
  launch_gemm(stream, 0, forecast, 256, in(IN_IFC_W1), 256, 1, in(IN_IFC_B1),
              H2, 256, 1, nullptr, nullptr, 256, 1024, 256, 256, 1.f, 0.f, 1,
              0.01f);
  launch_gemm(stream, 2, H2, 256, in(IN_IFC_W2), 41, 1, in(IN_IFC_B2), out3, 41,
              1, nullptr, nullptr, 256, 1024, 41, 256, 1.f, 0.f, 0, 0.f);
}